// PointPillarLoss_33251636805892
// MI455X (gfx1250) — compile-verified
//
#include <hip/hip_runtime.h>
#include <math.h>

// PointPillar loss: pure gather + reduction. Latency-bound (~16KB payload, ~4e4 FLOPs).
// One workgroup of 1024 threads (32 wave32s) does everything in a single pass.

#define FM_H 500
#define FM_W 440
#define NB   8      // batch
#define NBOX 50
#define NBG  100

__device__ __forceinline__ float warp_sum32(float v) {
    // wave32 butterfly reduction (ds_swizzle / permlane path on gfx1250)
    #pragma unroll
    for (int o = 16; o > 0; o >>= 1)
        v += __shfl_xor(v, o, 32);
    return v;
}

__device__ __forceinline__ float smooth_l1(float x) {
    float a = fabsf(x);
    return (a < 1.0f) ? (0.5f * x * x) : (a - 0.5f);
}

__device__ __forceinline__ float focal(float p) {
    float omp = 1.0f - p;
    return -logf(p) * 0.25f * omp * omp;   // ALPHA=0.25, GAMMA=2
}

__global__ __launch_bounds__(1024)
void pointpillar_loss_kernel(const int*   __restrict__ reg_t,    // (8,50,2)  int32
                             const int*   __restrict__ bg_t,     // (8,100,3) int32
                             const float* __restrict__ gt,       // (8,50,4)
                             const float* __restrict__ loc,      // (8,2,3,H,W)
                             const float* __restrict__ size_,    // (8,2,3,H,W)
                             const float* __restrict__ clf,      // (8,2,3,H,W)
                             const float* __restrict__ anchor_wh,// (2,)
                             float*       __restrict__ out)
{
    const int t = threadIdx.x;
    const long plane   = (long)FM_H * FM_W;   // 220000 elems
    const long bstride = 2L * 3L * plane;     // batch stride in loc/size/clf

    float s_bg = 0.f, s_dx = 0.f, s_dy = 0.f, s_dw = 0.f, s_dl = 0.f, s_car = 0.f;

    // ---- background focal terms: 800 gathers of clf[b,0,0,y,x] ----
    if (t < NB * NBG) {
        const int b = t / NBG, j = t % NBG;
        const int base = (b * NBG + j) * 3;
        const int x = bg_t[base + 1];
        const int y = bg_t[base + 2];
        const float p = clf[(long)b * bstride + (long)y * FM_W + x];
        s_bg = focal(p);
    }

    // ---- per-box terms: 400 boxes ----
    if (t < NB * NBOX) {
        const int b = t / NBOX, j = t % NBOX;
        const int ibase = (b * NBOX + j) * 2;
        const int xi = reg_t[ibase + 0];
        const int yi = reg_t[ibase + 1];
        const bool valid = (xi < FM_W) && (yi < FM_H);
        const int xc = min(max(xi, 0), FM_W - 1);
        const int yc = min(max(yi, 0), FM_H - 1);
        const long p00 = (long)yc * FM_W + xc;
        const long bb  = (long)b * bstride;

        const float* gb = gt + (b * NBOX + j) * 4;
        const float g0 = gb[0], g1 = gb[1], g2 = gb[2], g3 = gb[3];
        const float w_gt = g3 - g1;
        const float l_gt = g2 - g0;
        const float x_gt = g0 + 0.5f * w_gt;
        const float y_gt = g1 - 0.5f * l_gt;

        const float aw = anchor_wh[0], al = anchor_wh[1];
        const float inv_da = 1.0f / sqrtf(aw * aw + al * al);

        const float x_pred = valid ? loc[bb + p00]         : 0.f;
        const float y_pred = valid ? loc[bb + plane + p00] : 0.f;
        const float dx = valid ? (x_gt - x_pred) * inv_da : 0.f;
        const float dy = valid ? (y_gt - y_pred) * inv_da : 0.f;

        const float sw = fabsf(size_[bb + p00]);
        const float sl = fabsf(size_[bb + plane + p00]);
        const bool mw = valid && (w_gt != 0.f);
        const bool ml = valid && (l_gt != 0.f);
        const float dw = mw ? logf(w_gt / sw) : 0.f;
        const float dl = ml ? logf(l_gt / sl) : 0.f;

        const float p_car = clf[bb + plane + p00];   // clf[b,0,1]
        s_car = valid ? focal(p_car) : 0.f;

        s_dx = smooth_l1(dx);
        s_dy = smooth_l1(dy);
        s_dw = smooth_l1(dw);
        s_dl = smooth_l1(dl);
    }

    // ---- reduction: wave32 butterflies, then one LDS hop over 32 waves ----
    s_bg  = warp_sum32(s_bg);
    s_dx  = warp_sum32(s_dx);
    s_dy  = warp_sum32(s_dy);
    s_dw  = warp_sum32(s_dw);
    s_dl  = warp_sum32(s_dl);
    s_car = warp_sum32(s_car);

    __shared__ float lds[32][6];
    const int lane = t & 31;
    const int wave = t >> 5;
    if (lane == 0) {
        lds[wave][0] = s_bg;  lds[wave][1] = s_dx;  lds[wave][2] = s_dy;
        lds[wave][3] = s_dw;  lds[wave][4] = s_dl;  lds[wave][5] = s_car;
    }
    __syncthreads();

    if (wave == 0) {
        float v0 = lds[lane][0];
        float v1 = lds[lane][1];
        float v2 = lds[lane][2];
        float v3 = lds[lane][3];
        float v4 = lds[lane][4];
        float v5 = lds[lane][5];
        v0 = warp_sum32(v0); v1 = warp_sum32(v1); v2 = warp_sum32(v2);
        v3 = warp_sum32(v3); v4 = warp_sum32(v4); v5 = warp_sum32(v5);
        if (lane == 0) {
            const float bg_focal  = v0 * (1.0f / (NB * NBG));          // mean over 800
            const float loc_loss  = (v1 + v2 + v3 + v4) * (1.0f / (NB * NBOX)); // 4 means over 400
            const float car_focal = v5 * (1.0f / (NB * NBOX));         // sum / 400
            // BETA_LOC=2.0, BETA_CLS=1.0
            out[0] = 2.0f * loc_loss + (bg_focal + car_focal);
        }
    }
}

extern "C" void kernel_launch(void* const* d_in, const int* in_sizes, int n_in,
                              void* d_out, int out_size, void* d_ws, size_t ws_size,
                              hipStream_t stream) {
    (void)in_sizes; (void)n_in; (void)out_size; (void)d_ws; (void)ws_size;
    const int*   reg_t  = (const int*)  d_in[0];  // regression_targets (8,50,2)
    const int*   bg_t   = (const int*)  d_in[1];  // bg_targets (8,100,3)
    const float* gt     = (const float*)d_in[2];  // gt_boxes_tensor (8,50,4)
    const float* loc    = (const float*)d_in[3];  // (8,2,3,500,440)
    const float* size_  = (const float*)d_in[4];
    const float* clf    = (const float*)d_in[5];
    // d_in[6..8] = occupancy/angle/heading: unused by the reference loss
    const float* anchor = (const float*)d_in[9];  // (2,)
    float* out = (float*)d_out;

    pointpillar_loss_kernel<<<1, 1024, 0, stream>>>(reg_t, bg_t, gt, loc, size_, clf,
                                                    anchor, out);
}